// RiemannianAutoencoder_11605001634118
// MI455X (gfx1250) — compile-verified
//
#include <hip/hip_runtime.h>
#include <math.h>

// ---------------- static problem config (mirrors reference) ----------------
#define N_BATCH   8192
#define T_NSTEPS  50
#define MSTAR     1728            // 12*12*12
#define KSTEPS    432             // MSTAR / 4 (WMMA K=4 steps)
#define WB_FLOATS (KSTEPS * 64)   // pre-swizzled B fragments: [432][32 lanes][2]
#define PI_F      3.14159265358979323846f

// ---------------- launch geometry ----------------
#define THREADS_PB 128            // 4 waves of 32
#define WAVES_PB   4
#define PARTS_PW   16             // particles per wave (WMMA M)
#define PARTS_PB   (WAVES_PB * PARTS_PW)   // 64
#define NBLOCKS    (N_BATCH / PARTS_PB)    // 128

// ---------------- per-wave LDS layout (floats) ----------------
#define PE_OFF   0                 // [16][3] eval positions
#define S_OFF    48                // [16][3][12] scaled sin
#define C_OFF    624               // [16][3][12] scaled cos * w
#define OUT_OFF  1200              // [16][4 feat][6 comp] GEMM results
#define PW_FLOATS 1584
#define SMEM_FLOATS (WB_FLOATS + WAVES_PB * PW_FLOATS)   // 33984 -> 135936 B

typedef __attribute__((ext_vector_type(2))) float v2f;
typedef __attribute__((ext_vector_type(8))) float v8f;

__device__ __forceinline__ v8f wmma4(v2f a, v2f b, v8f c) {
  // D(16x16 f32) = A(16x4 f32) * B(4x16 f32) + C
  return __builtin_amdgcn_wmma_f32_16x16x4_f32(
      /*neg_a=*/false, a, /*neg_b=*/false, b,
      /*c_mod=*/(short)0, c, /*reuse_a=*/false, /*reuse_b=*/false);
}

// ============================================================================
// Kernel 1: build L and the pre-swizzled, zero-padded weight matrix
// wb[e]: e = t*64 + lane*2 + v  ->  B-fragment element for WMMA k-step t,
// lane "lane", VGPR v.  col d = lane&15 (>=6 zero-padded),
// row m = 4*t + (lane<16 ? v : v+2).
// ============================================================================
__global__ void geo_precompute(const float* __restrict__ beta,
                               const float* __restrict__ ls,
                               const float* __restrict__ basis,
                               float* __restrict__ wb,
                               float* __restrict__ Lout) {
  __shared__ float Ls[3];
  const int tid = threadIdx.x;
  if (tid < 3) {
    float m = 0.f;
    for (int r = 0; r < 256; ++r) m = fmaxf(m, fabsf(basis[r * 3 + tid]));
    const float L = 1.5f * m;
    Ls[tid] = L;
    Lout[tid] = L;
  }
  __syncthreads();
  const float c34 = powf(2.0f * PI_F, 0.75f);       // (2*pi)^(3/4)
  const float h0 = PI_F / (2.f * Ls[0]);
  const float h1 = PI_F / (2.f * Ls[1]);
  const float h2 = PI_F / (2.f * Ls[2]);
  for (int e = tid; e < WB_FLOATS; e += blockDim.x) {
    const int t    = e >> 6;
    const int rem  = e & 63;
    const int lane = rem >> 1;
    const int v    = rem & 1;
    const int d    = lane & 15;
    const int m    = 4 * t + ((lane & 16) ? (v + 2) : v);
    float val = 0.f;
    if (d < 6) {
      const int kk = m % 12, j = (m / 12) % 12, i = m / 144;
      const float o0 = h0 * (float)(i + 1);
      const float o1 = h1 * (float)(j + 1);
      const float o2 = h2 * (float)(kk + 1);
      const float l0 = ls[d * 3 + 0], l1 = ls[d * 3 + 1], l2 = ls[d * 3 + 2];
      const float ex = -0.25f * (o0 * o0 * l0 * l0 + o1 * o1 * l1 * l1 +
                                 o2 * o2 * l2 * l2);
      const float spd = c34 * sqrtf(l0 * l1 * l2) * expf(ex);
      val = beta[d * MSTAR + m] * spd;
    }
    wb[e] = val;
  }
}

// ============================================================================
// Kernel 2: RK4 geodesic integration.  One wave = 16 particles; HSGP
// contraction done with V_WMMA_F32_16X16X4_F32 (4 accumulators: phi,d0,d1,d2).
// ============================================================================
__global__ void __launch_bounds__(THREADS_PB)
geo_integrate(const float* __restrict__ pos0, const float* __restrict__ vel0,
              const float* __restrict__ wb, const float* __restrict__ Lg,
              float* __restrict__ out) {
  extern __shared__ float smem[];
  const int tid  = threadIdx.x;
  const int lane = tid & 31;
  const int wave = tid >> 5;
  const int p    = lane & 15;                        // particle slot in wave
  const int b    = blockIdx.x * PARTS_PB + wave * PARTS_PW + p;

  float* wbl  = smem;                                // [432][32][2] B frags
  float* pw   = smem + WB_FLOATS + wave * PW_FLOATS;
  float* PE   = pw + PE_OFF;
  float* Sb   = pw + S_OFF;
  float* Cb   = pw + C_OFF;
  float* OUTb = pw + OUT_OFF;

  // stage W fragments into LDS (once per block)
  for (int e = tid; e < WB_FLOATS; e += THREADS_PB) wbl[e] = wb[e];

  const float Lv[3]  = {Lg[0], Lg[1], Lg[2]};
  const float hw[3]  = {PI_F / (2.f * Lv[0]), PI_F / (2.f * Lv[1]),
                        PI_F / (2.f * Lv[2])};
  const float isq[3] = {1.f / sqrtf(Lv[0]), 1.f / sqrtf(Lv[1]),
                        1.f / sqrtf(Lv[2])};

  float P[3], V[3];
#pragma unroll
  for (int d = 0; d < 3; ++d) {
    P[d] = pos0[b * 3 + d];
    V[d] = vel0[b * 3 + d];
  }

  // trajectory at t=0
  if (lane < 16) {
#pragma unroll
    for (int d = 0; d < 3; ++d) {
      out[(size_t)(0 * N_BATCH + b) * 3 + d] = P[d];           // pos, t=0
      out[(size_t)(1 * N_BATCH + b) * 3 + d] = V[d];           // vel, t=0
    }
  }
  __syncthreads();

  const float dt = 1.0f / (float)(T_NSTEPS - 1);
  const int cidx[3][3] = {{0, 1, 3}, {1, 2, 4}, {3, 4, 5}};    // sym index map
  const int khalf = (lane >> 4) << 1;                          // A-tile K: 0 or 2

  for (int step = 1; step < T_NSTEPS; ++step) {
    float pe[3] = {P[0], P[1], P[2]};
    float ve[3] = {V[0], V[1], V[2]};
    float sP[3] = {0.f, 0.f, 0.f}, sV[3] = {0.f, 0.f, 0.f};

#pragma unroll 1
    for (int sstep = 0; sstep < 4; ++sstep) {
      // ---- publish eval positions ----
      if (lane < 16) {
        PE[p * 3 + 0] = pe[0];
        PE[p * 3 + 1] = pe[1];
        PE[p * 3 + 2] = pe[2];
      }
      __syncthreads();

      // ---- cooperative trig: 16 parts * 3 dims * 12 freqs = 576 sincos ----
#pragma unroll 1
      for (int n = 0; n < 18; ++n) {
        const int idx = lane + 32 * n;          // 0..575
        const int pp  = idx / 36;
        const int r   = idx - 36 * pp;
        const int dd  = r / 12;
        const int kk  = r - 12 * dd;
        const float w   = hw[dd] * (float)(kk + 1);
        const float arg = w * (PE[pp * 3 + dd] + Lv[dd]);
        float s, c;
        __sincosf(arg, &s, &c);
        Sb[(pp * 3 + dd) * 12 + kk] = isq[dd] * s;
        Cb[(pp * 3 + dd) * 12 + kk] = isq[dd] * c * w;
      }
      __syncthreads();

      // ---- HSGP contraction: 4 GEMMs [16 x 1728] x [1728 x 16(6)] ----
      v8f accP  = {0.f, 0.f, 0.f, 0.f, 0.f, 0.f, 0.f, 0.f};
      v8f accD0 = accP, accD1 = accP, accD2 = accP;
      const float* Sp0 = Sb + p * 36;           // S[p][0][*]
      const float* Cp0 = Cb + p * 36;
      const float* Sp2 = Sp0 + 24;              // S[p][2][*]
      const float* Cp2 = Cp0 + 24;
#pragma unroll 1
      for (int i = 0; i < 12; ++i) {
        const float s0 = Sp0[i];
        const float c0 = Cp0[i];
#pragma unroll 1
        for (int j = 0; j < 12; ++j) {
          const float s1 = Sp0[12 + j];
          const float c1 = Cp0[12 + j];
          const float a  = s0 * s1;             // phi / d2 prefix
          const float a0 = c0 * s1;             // d0 prefix
          const float a1 = s0 * c1;             // d1 prefix
          const int tbase = (i * 12 + j) * 3;
#pragma unroll
          for (int kk3 = 0; kk3 < 3; ++kk3) {
            const int k0 = kk3 * 4 + khalf;
            const v2f s2 = *(const v2f*)(Sp2 + k0);
            const v2f c2 = *(const v2f*)(Cp2 + k0);
            const v2f Bf = *(const v2f*)(wbl + (tbase + kk3) * 64 + lane * 2);
            const v2f Ap = {a * s2.x, a * s2.y};
            const v2f A0 = {a0 * s2.x, a0 * s2.y};
            const v2f A1 = {a1 * s2.x, a1 * s2.y};
            const v2f A2 = {a * c2.x, a * c2.y};
            accP  = wmma4(Ap, Bf, accP);
            accD0 = wmma4(A0, Bf, accD0);
            accD1 = wmma4(A1, Bf, accD1);
            accD2 = wmma4(A2, Bf, accD2);
          }
        }
      }

      // ---- C-tile (M=particle, N=component) -> LDS ----
      {
        const int n   = lane & 15;
        const int mhi = (lane >> 4) * 8;
        if (n < 6) {
#pragma unroll
          for (int r = 0; r < 8; ++r) {
            const int mp = mhi + r;
            OUTb[(mp * 4 + 0) * 6 + n] = accP[r];
            OUTb[(mp * 4 + 1) * 6 + n] = accD0[r];
            OUTb[(mp * 4 + 2) * 6 + n] = accD1[r];
            OUTb[(mp * 4 + 3) * 6 + n] = accD2[r];
          }
        }
      }
      __syncthreads();

      // ---- Christoffel + acceleration (all lanes; halves duplicated) ----
      float gf[6], dgf[6][3];
#pragma unroll
      for (int c = 0; c < 6; ++c) {
        gf[c] = OUTb[(p * 4 + 0) * 6 + c];
#pragma unroll
        for (int r = 0; r < 3; ++r) dgf[c][r] = OUTb[(p * 4 + 1 + r) * 6 + c];
      }
      const float g00 = gf[0] + 1.f, g01 = gf[1], g11 = gf[2] + 1.f;
      const float g02 = gf[3], g12 = gf[4], g22 = gf[5] + 1.f;
      const float A00 = g11 * g22 - g12 * g12;
      const float A01 = g02 * g12 - g01 * g22;
      const float A02 = g01 * g12 - g02 * g11;
      const float det = g00 * A00 + g01 * A01 + g02 * A02;
      const float rd  = 1.f / det;
      const float ginv[3][3] = {
          {A00 * rd, A01 * rd, A02 * rd},
          {A01 * rd, (g00 * g22 - g02 * g02) * rd, (g01 * g02 - g00 * g12) * rd},
          {A02 * rd, (g01 * g02 - g00 * g12) * rd, (g00 * g11 - g01 * g01) * rd}};
      float u[3], wl[3];
#pragma unroll
      for (int l = 0; l < 3; ++l) {
        float su = 0.f, sw = 0.f;
#pragma unroll
        for (int ii = 0; ii < 3; ++ii) {
#pragma unroll
          for (int jj = 0; jj < 3; ++jj) {
            const float vv = ve[ii] * ve[jj];
            su += vv * dgf[cidx[l][jj]][ii];   // v^i v^j d_i g_{lj}
            sw += vv * dgf[cidx[ii][jj]][l];   // v^i v^j d_l g_{ij}
          }
        }
        u[l] = su;
        wl[l] = sw;
      }
      const float bb[3] = {u[0] - 0.5f * wl[0], u[1] - 0.5f * wl[1],
                           u[2] - 0.5f * wl[2]};
      float acc[3];
#pragma unroll
      for (int k = 0; k < 3; ++k)
        acc[k] = -(ginv[k][0] * bb[0] + ginv[k][1] * bb[1] + ginv[k][2] * bb[2]);

      // ---- RK4 accumulate / advance eval state ----
      const float wgt = (sstep == 0 || sstep == 3) ? 1.f : 2.f;
#pragma unroll
      for (int d = 0; d < 3; ++d) {
        sP[d] += wgt * ve[d];
        sV[d] += wgt * acc[d];
      }
      if (sstep < 3) {
        const float cf = (sstep == 2) ? dt : 0.5f * dt;
#pragma unroll
        for (int d = 0; d < 3; ++d) {
          pe[d] = P[d] + cf * ve[d];   // uses pre-update ve
          ve[d] = V[d] + cf * acc[d];
        }
      }
      __syncthreads();   // protect OUTb/PE before next substep rewrites
    }

#pragma unroll
    for (int d = 0; d < 3; ++d) {
      P[d] += (dt / 6.f) * sP[d];
      V[d] += (dt / 6.f) * sV[d];
    }
    if (lane < 16) {
#pragma unroll
      for (int d = 0; d < 3; ++d) {
        out[(size_t)((step * 2 + 0) * N_BATCH + b) * 3 + d] = P[d];
        out[(size_t)((step * 2 + 1) * N_BATCH + b) * 3 + d] = V[d];
      }
    }
  }
}

// ============================================================================
extern "C" void kernel_launch(void* const* d_in, const int* in_sizes, int n_in,
                              void* d_out, int out_size, void* d_ws,
                              size_t ws_size, hipStream_t stream) {
  const float* pos0  = (const float*)d_in[0];   // [8192,3]
  const float* vel0  = (const float*)d_in[1];   // [8192,3]
  const float* beta  = (const float*)d_in[2];   // [6,1728]
  const float* ls    = (const float*)d_in[3];   // [6,3]
  const float* basis = (const float*)d_in[4];   // [256,3]
  float* out = (float*)d_out;                   // [50,2,8192,3]

  float* wb   = (float*)d_ws;                   // WB_FLOATS floats
  float* Lout = wb + WB_FLOATS;                 // 3 floats

  geo_precompute<<<1, 256, 0, stream>>>(beta, ls, basis, wb, Lout);

  const size_t smem = (size_t)SMEM_FLOATS * sizeof(float);  // ~136 KB
  (void)hipFuncSetAttribute((const void*)geo_integrate,
                            hipFuncAttributeMaxDynamicSharedMemorySize,
                            (int)smem);
  geo_integrate<<<NBLOCKS, THREADS_PB, smem, stream>>>(pos0, vel0, wb, Lout,
                                                       out);
}